// HarmonicOscillators_53506702574000
// MI455X (gfx1250) — compile-verified
//
#include <hip/hip_runtime.h>
#include <hip/hip_bf16.h>

// HarmonicOscillators for MI455X (gfx1250), wave32.
//
// Roofline: 4MB out / 1.7MB in @ 23.3TB/s => ~0.25us memory floor; 65M
// sin+mac terms => compute-shaped but tiny. Strategy:
//   stage 1: per-batch f64 scan of closed-form per-block phase increments
//            (kills the 64000-long serial cumsum; 16x400 values).
//   stage 2: one 320-thread WG (10 waves) per (batch, control-block).
//            Each wave computes a 16x16 tile S[t,n] = sum_k sin(k*phi_t)*
//            alpha[knot_n,k] via 16 chained V_WMMA_F32_16X16X4_F32 (f32
//            precision == reference). Output y = ampu*((1-w)S[cA]+w*S[cB]).
// Phase kept in cycles; v_sin_f32 (sin of 2*pi*x) fed with frac'd input.

#define SR_D   16000.0
#define TBLK   160
#define NBLK   400
#define NB     16
#define NPART  64
#define TWO_PI_F 6.2831853071795864769f

typedef __attribute__((ext_vector_type(2))) float v2f;
typedef __attribute__((ext_vector_type(8))) float v8f;

// ---------------- Stage 1: per-block phase base (cycles), f64 scan --------
__global__ __launch_bounds__(128)
void HarmOsc_phase_scan(const float* __restrict__ f0, double* __restrict__ q0) {
    __shared__ double delta[NBLK];
    const int b = blockIdx.x;
    const float* f0b = f0 + b * NBLK;
    for (int j = threadIdx.x; j < NBLK; j += blockDim.x) {
        const int jm = max(j - 1, 0), jp = min(j + 1, NBLK - 1);
        const double gm = (double)f0b[jm] / SR_D;
        const double g0 = (double)f0b[j]  / SR_D;
        const double gp = (double)f0b[jp] / SR_D;
        // sum of 160 linearly-upsampled f0/sr samples of block j (closed form)
        delta[j] = 20.0 * gm + 120.0 * g0 + 20.0 * gp;
    }
    __syncthreads();
    if (threadIdx.x == 0) {
        double acc = -(double)f0b[0] / SR_D;   // reference subtracts f0u[0]
        for (int j = 0; j < NBLK; ++j) { q0[b * NBLK + j] = acc; acc += delta[j]; }
    }
}

// ---------------- Stage 2: synthesis --------------------------------------
__global__ __launch_bounds__(320)
void HarmOsc_synth(const float* __restrict__ amp, const float* __restrict__ alpha,
                   const float* __restrict__ f0, const double* __restrict__ q0,
                   float* __restrict__ out) {
    __shared__ float aK[3 * NPART];        // alpha knot rows jm1/j/jp1, [n][k]
    __shared__ float Sbuf[10][16][3];      // per-tile GEMM results, 3 used cols

    const int j = blockIdx.x;              // control block 0..399
    const int b = blockIdx.y;              // batch
    const int jm = max(j - 1, 0), jp = min(j + 1, NBLK - 1);
    const int tid = threadIdx.x;

    const float* alb = alpha + (size_t)b * NBLK * NPART;
    if (tid < 3 * NPART) {
        const int n = tid >> 6, k = tid & 63;
        const int row = (n == 0) ? jm : ((n == 1) ? j : jp);
        aK[tid] = alb[row * NPART + k];
    }
    // prefetch next block's alpha row into cache (global_prefetch_b8)
    __builtin_prefetch(alb + (size_t)min(j + 2, NBLK - 1) * NPART, 0, 0);

    const float gm  = f0[b * NBLK + jm] * (1.0f / 16000.0f);
    const float g0f = f0[b * NBLK + j]  * (1.0f / 16000.0f);
    const float gp  = f0[b * NBLK + jp] * (1.0f / 16000.0f);
    const float am  = amp[b * NBLK + jm];
    const float a0  = amp[b * NBLK + j];
    const float ap  = amp[b * NBLK + jp];
    const double qbase = q0[b * NBLK + j];
    __syncthreads();

    const int wave = tid >> 5, lane = tid & 31;
    const int r = lane & 15, hi = lane >> 4;
    const int i = wave * 16 + r;           // sample within block, 0..159
    const int t = j * TBLK + i;

    // per-sample upsample params (same f32 formula as the reference)
    const float src = fmaxf((float)(t) * (1.0f / 160.0f) + 0.5f * (1.0f / 160.0f) - 0.5f, 0.0f);
    const int   x0  = (int)src;
    const float w   = src - (float)x0;
    const int   x1  = min(x0 + 1, NBLK - 1);
    const int   cA  = x0 - j + 1;          // in {0,1,2}
    const int   cB  = x1 - j + 1;          // in {0,1,2}
    const float gA  = (cA == 0) ? gm : ((cA == 1) ? g0f : gp);
    const float gB  = (cB == 0) ? gm : ((cB == 1) ? g0f : gp);
    const float f0u = gA * (1.0f - w) + gB * w;     // for anti-alias mask

    // inclusive within-block phase partial sum (cycles), closed form, f64
    const double gmD = (double)gm, g0D = (double)g0f, gpD = (double)gp;
    double P;
    if (i < 80) {
        const double n1 = (double)(i + 1);
        P = n1 * gmD + (g0D - gmD) * (0.5 * n1 + n1 * n1 * (1.0 / 320.0));
    } else {
        const double rr = (double)(i - 79);
        P = 20.0 * gmD + 60.0 * g0D + rr * g0D + (gpD - g0D) * (rr * rr * (1.0 / 320.0));
    }
    const double qd = qbase + P;
    const float  qf = (float)(qd - floor(qd));      // phase in [0,1) cycles

    // ---- 16 x16x4 f32 WMMA steps: D += A(sin) x B(alpha knots) ----
    // A 16x4: lane(r,hi) supplies K = 4s+2hi+{0,1} of row r.
    // B 4x16: lane(n=r,hi) supplies K = 4s+2hi+{0,1} of column n (n<3 live).
    const int  rB = min(r, 2);
    const bool bLive = (r < 3);
    v8f c = {};
    #pragma unroll
    for (int s = 0; s < 16; ++s) {
        const int kb = 4 * s + 2 * hi;
        v2f a, bb;
        #pragma unroll
        for (int e = 0; e < 2; ++e) {
            const int   h  = kb + e + 1;            // harmonic number 1..64
            const float x  = (float)h * qf;
            const float xr = x - floorf(x);         // [0,1) revolutions
            float sv = __builtin_amdgcn_sinf(xr);   // v_sin_f32: sin(2*pi*xr)
            sv = ((float)h * f0u < 0.5f) ? sv : 0.0f;   // anti-alias gate
            a[e]  = sv;
            const float bv = aK[rB * NPART + kb + e];
            bb[e] = bLive ? bv : 0.0f;
        }
        c = __builtin_amdgcn_wmma_f32_16x16x4_f32(false, a, false, bb,
                                                  (short)0, c, false, false);
    }

    // C/D layout: lane holds column n=lane&15, rows M = hi*8 + v (v=0..7)
    if (r < 3) {
        #pragma unroll
        for (int v = 0; v < 8; ++v) Sbuf[wave][hi * 8 + v][r] = c[v];
    }
    __syncthreads();

    // ---- final interp + amplitude + store ----
    if (tid < TBLK) {
        const int   i2   = tid;
        const int   t2   = j * TBLK + i2;
        const float src2 = fmaxf((float)(t2) * (1.0f / 160.0f) + 0.5f * (1.0f / 160.0f) - 0.5f, 0.0f);
        const int   y0   = (int)src2;
        const float w2   = src2 - (float)y0;
        const int   y1   = min(y0 + 1, NBLK - 1);
        const int   dA   = y0 - j + 1;
        const int   dB   = y1 - j + 1;
        const float aA   = (dA == 0) ? am : ((dA == 1) ? a0 : ap);
        const float aB   = (dB == 0) ? am : ((dB == 1) ? a0 : ap);
        const float ampu = aA * (1.0f - w2) + aB * w2;
        const int tile = i2 >> 4, rr2 = i2 & 15;
        const float Sa = Sbuf[tile][rr2][dA];
        const float Sb = Sbuf[tile][rr2][dB];
        out[(size_t)b * (NBLK * TBLK) + t2] = ampu * ((1.0f - w2) * Sa + w2 * Sb);
    }
}

extern "C" void kernel_launch(void* const* d_in, const int* in_sizes, int n_in,
                              void* d_out, int out_size, void* d_ws, size_t ws_size,
                              hipStream_t stream) {
    const float* amp   = (const float*)d_in[0];   // [16,400,1]
    const float* alpha = (const float*)d_in[1];   // [16,400,64]
    const float* f0    = (const float*)d_in[2];   // [16,400,1]
    float* out = (float*)d_out;                   // [16,64000]
    double* q0 = (double*)d_ws;                   // 16*400 doubles = 51.2 KB

    HarmOsc_phase_scan<<<dim3(NB), 128, 0, stream>>>(f0, q0);
    HarmOsc_synth<<<dim3(NBLK, NB), 320, 0, stream>>>(amp, alpha, f0, q0, out);
}